// RotatedRoIAlign_944892805763
// MI455X (gfx1250) — compile-verified
//
#include <hip/hip_runtime.h>
#include <math.h>

// Rotated RoI Align for MI455X (gfx1250).
// Bandwidth-bound irregular gather: taps are data-dependent per sample point,
// so there is no GEMM structure => no WMMA (forcing it would inflate traffic).
// Strategy per MI455X specs: feature pyramid (178 MB) fits the 192 MB L2, so
// gathers are L2-resident after first touch; output (51 MB) is streamed with
// non-temporal b128 stores so it never evicts features. 16 gathers in flight
// per thread (float4 output groups), packed LDS point-descriptors
// (ds_load_b128/b64), global_prefetch_b8 for next taps, wave32 blocks of
// 224 threads (7 waves) => 14 exact iterations per thread, no tail.

#define OUT_S   7
#define NPTS    49          // 7*7
#define BLOCK   224         // 3136 float4s per roi tile = 14 * 224 exactly

typedef float v4f __attribute__((ext_vector_type(4)));

struct __align__(16) Taps { int o00, o01, o10, o11; };
struct __align__(8)  Wgt  { float wx, wy; };

__global__ __launch_bounds__(BLOCK) void rroi_align_gfx1250(
    const float* __restrict__ f0, const float* __restrict__ f1,
    const float* __restrict__ f2, const float* __restrict__ f3,
    const float* __restrict__ rois, const int* __restrict__ levels,
    float* __restrict__ out, int B, int N, int C)
{
    __shared__ Taps s_tap[NPTS];
    __shared__ Wgt  s_w[NPTS];

    const int bn = blockIdx.x;            // roi index: b*N + n
    const int b  = bn / N;
    const int lv = levels[bn];            // uniform per block

    const float* feat;
    int W;
    switch (lv) {
        case 0:  feat = f0; W = 256; break;
        case 1:  feat = f1; W = 128; break;
        case 2:  feat = f2; W = 64;  break;
        default: feat = f3; W = 32;  break;
    }
    const int   H  = W;
    const float sc = 1.0f / (float)(4 << lv);   // 0.25,...,0.03125 (exact pow2)

    // ---- per-roi sample-point setup: first 49 lanes ----
    if (threadIdx.x < NPTS) {
        const int p = threadIdx.x;
        const int i = p / OUT_S;
        const int j = p - i * OUT_S;

        const float* r = rois + (size_t)bn * 6;
        const float x = r[0], y = r[1], w = r[2], h = r[3], a = r[4], bb = r[5];

        // parallelogram -> rect (matches reference op order)
        const float d1  = sqrtf(4.0f * a * a + h * h);
        const float d2  = sqrtf(w * w + 4.0f * bb * bb);
        const float eps = 1e-6f;
        const bool  cond = d1 > d2;
        const float sA = d1 / (d2 + eps);
        const float sB = d2 / (d1 + eps);
        const float V0x = cond ? (x + a)             : (x + sB * a);
        const float V0y = cond ? (y + 0.5f * h)      : (y + sB * 0.5f * h);
        const float V1x = cond ? (x + sA * 0.5f * w) : (x + 0.5f * w);
        const float V1y = cond ? (y + sA * bb)       : (y + bb);
        const float V2x = cond ? (x - a)             : (x - sB * a);
        const float V2y = cond ? (y - 0.5f * h)      : (y - sB * 0.5f * h);

        const float dx10 = V1x - V0x, dy10 = V1y - V0y;
        const float h_rect = sqrtf(dx10 * dx10 + dy10 * dy10);
        const float dx21 = V2x - V1x, dy21 = V2y - V1y;
        const float w_rect = sqrtf(dx21 * dx21 + dy21 * dy21);

        const float HPI = 1.57079632679489661923f;
        float th = atan2f(V1y - V2y, V1x - V2x);
        th = fminf(fmaxf(th, -HPI), HPI);

        const float x_r = x * sc, y_r = y * sc;
        const float w_r = w_rect * sc, h_r = h_rect * sc;

        const float linj = -0.5f + (float)j * (1.0f / 6.0f);
        const float lini = -0.5f + (float)i * (1.0f / 6.0f);
        const float gx = w_r * linj;
        const float gy = h_r * lini;

        float sth, cth;
        sincosf(th, &sth, &cth);
        const float rx =  gx * cth + gy * sth;
        const float ry = -gx * sth + gy * cth;
        const float px = (rx + x_r) * sc;   // reference scales twice; replicate
        const float py = (ry + y_r) * sc;

        // grid-sample style unnormalize + clamp (literal reference formula)
        const float nx = px / (float)(W - 1) * 2.0f - 1.0f;
        const float ny = py / (float)(H - 1) * 2.0f - 1.0f;
        float ux = ((nx + 1.0f) * (float)W - 1.0f) * 0.5f;
        float uy = ((ny + 1.0f) * (float)H - 1.0f) * 0.5f;
        ux = fminf(fmaxf(ux, 0.0f), (float)(W - 1));
        uy = fminf(fmaxf(uy, 0.0f), (float)(H - 1));

        const float x0 = floorf(ux), y0 = floorf(uy);
        const int x0i = (int)x0, y0i = (int)y0;
        const int x1i = (x0i + 1 < W) ? (x0i + 1) : (W - 1);
        const int y1i = (y0i + 1 < H) ? (y0i + 1) : (H - 1);

        Taps t;
        t.o00 = y0i * W + x0i;
        t.o01 = y0i * W + x1i;
        t.o10 = y1i * W + x0i;
        t.o11 = y1i * W + x1i;
        s_tap[p] = t;
        Wgt wg; wg.wx = ux - x0; wg.wy = uy - y0;
        s_w[p] = wg;
    }
    __syncthreads();

    // ---- gather + bilinear lerp: 4 consecutive outputs per thread/iter ----
    const size_t plane = (size_t)H * (size_t)W;
    const float* __restrict__ fb = feat + (size_t)b * (size_t)C * plane;
    v4f* __restrict__ ob4 =
        (v4f*)(out + (size_t)bn * (size_t)(C * NPTS));   // 50176 B tile, 16B aligned

    const int NQ = (C * NPTS) / 4;          // 3136 = 14 * 224 : no tail
    // starting (channel, point) of this thread's first 4-element group
    int c0 = (4 * (int)threadIdx.x) / NPTS;
    int p0 = 4 * (int)threadIdx.x - c0 * NPTS;

    for (int q = threadIdx.x; q < NQ; q += BLOCK) {
        // next group's start (stride 4*BLOCK = 896 = 18*49 + 14)
        int cn = c0 + 18, pn = p0 + 14;
        if (pn >= NPTS) { pn -= NPTS; cn += 1; }

        int cc = c0, pp = p0;
        v4f res;
        #pragma unroll
        for (int e = 0; e < 4; ++e) {
            const float* fc = fb + (size_t)cc * plane;
            const Taps o  = s_tap[pp];     // ds_load_b128
            const Wgt  wv = s_w[pp];       // ds_load_b64
            const float v00 = fc[o.o00];
            const float v01 = fc[o.o01];
            const float v10 = fc[o.o10];
            const float v11 = fc[o.o11];
            const float top = v00 * (1.0f - wv.wx) + v01 * wv.wx;
            const float bot = v10 * (1.0f - wv.wx) + v11 * wv.wx;
            res[e] = top * (1.0f - wv.wy) + bot * wv.wy;
            pp += 1;
            if (pp == NPTS) { pp = 0; cc += 1; }
        }

        // prefetch next group's corner tap cachelines (global_prefetch_b8)
        if (q + BLOCK < NQ) {
            const float* fcn = fb + (size_t)cn * plane;
            const Taps on = s_tap[pn];
            __builtin_prefetch(fcn + on.o00, 0, 3);
            __builtin_prefetch(fcn + on.o11, 0, 3);
        }

        // streaming NT b128 store: keep the feature pyramid resident in L2
        __builtin_nontemporal_store(res, &ob4[q]);

        c0 = cn; p0 = pn;
    }
}

extern "C" void kernel_launch(void* const* d_in, const int* in_sizes, int n_in,
                              void* d_out, int out_size, void* d_ws, size_t ws_size,
                              hipStream_t stream) {
    const float* f0     = (const float*)d_in[0];
    const float* f1     = (const float*)d_in[1];
    const float* f2     = (const float*)d_in[2];
    const float* f3     = (const float*)d_in[3];
    const float* rois   = (const float*)d_in[4];
    const int*   levels = (const int*)d_in[5];
    float*       out    = (float*)d_out;

    const int C  = 256;
    const int BN = in_sizes[5];                      // B*N
    const int B  = in_sizes[0] / (C * 256 * 256);    // feat0 = B*C*256*256
    const int N  = BN / B;

    hipLaunchKernelGGL(rroi_align_gfx1250, dim3(BN), dim3(BLOCK), 0, stream,
                       f0, f1, f2, f3, rois, levels, out, B, N, C);
}